// HeteroSAGE_31636729102833
// MI455X (gfx1250) — compile-verified
//
#include <hip/hip_runtime.h>

typedef __attribute__((ext_vector_type(2))) float v2f;
typedef __attribute__((ext_vector_type(8))) float v8f;

#define N_PAPER  100000
#define N_AUTHOR 50000
#define N_FIELD  10000
#define HIDDEN   128
#define OUT_DIM  16
#define ALPHA    0.1f
#define EPS      1e-4f

// ---------------- zero fill ----------------
__global__ void k_zero(float* __restrict__ p, long n) {
    long i = (long)blockIdx.x * blockDim.x + threadIdx.x;
    long stride = (long)gridDim.x * blockDim.x;
    for (; i < n; i += stride) p[i] = 0.0f;
}

// ---------------- C[M x 128] = relu(A[M x 128] @ W[128 x 128]) ----------------
// One block = 8 waves. Each block processes 16-row strips (grid-stride).
// Full W (64KB) + A strip (8KB) staged in LDS. Each wave computes one 16x16
// output tile with V_WMMA_F32_16X16X4_F32 over K=128 (32 WMMAs).
// In-place safe: the strip is fully read into LDS before C is written.
__global__ void __launch_bounds__(256) k_gemm_relu(const float* __restrict__ A,
                                                   const float* __restrict__ W,
                                                   float* __restrict__ C, int M) {
    __shared__ float sW[HIDDEN * HIDDEN];   // 64 KB
    __shared__ float sA[16 * HIDDEN];       // 8 KB

    for (int i = threadIdx.x; i < HIDDEN * HIDDEN; i += 256) sW[i] = W[i];

    const int nstrips = M >> 4;
    const int wave = threadIdx.x >> 5;
    const int lane = threadIdx.x & 31;
    const int n0   = wave << 4;       // this wave's N tile
    const int m16  = lane & 15;
    const int hi   = lane >> 4;       // 0: K=k,k+1   1: K=k+2,k+3

    for (int strip = blockIdx.x; strip < nstrips; strip += gridDim.x) {
        __syncthreads();  // protect sA (and sW on first iter)
        long base = (long)strip * 16 * HIDDEN;
        for (int i = threadIdx.x; i < 16 * HIDDEN; i += 256) sA[i] = A[base + i];
        __syncthreads();

        const float* aRow = &sA[m16 * HIDDEN + hi * 2];          // A[m][k + hi*2]
        const float* bCol = &sW[(hi * 2) * HIDDEN + n0 + m16];   // W[k + hi*2][n]

        v8f acc = {};
#pragma unroll
        for (int k = 0; k < HIDDEN; k += 4) {
            v2f af; af.x = aRow[k];           af.y = aRow[k + 1];
            v2f bf; bf.x = bCol[k * HIDDEN];  bf.y = bCol[k * HIDDEN + HIDDEN];
            acc = __builtin_amdgcn_wmma_f32_16x16x4_f32(
                false, af, false, bf, (short)0, acc, false, false);
        }

        // C/D layout: VGPR r -> (M = r + hi*8, N = n0 + m16)
#pragma unroll
        for (int r = 0; r < 8; r++) {
            float v = acc[r];
            v = v > 0.0f ? v : 0.0f;
            C[base + (long)(r + hi * 8) * HIDDEN + n0 + m16] = v;
        }
    }
}

// ---------------- scatter-add: one wave32 per edge ----------------
__global__ void k_scatter(const float* __restrict__ hsrc,
                          const int* __restrict__ src,
                          const int* __restrict__ dst,
                          float* __restrict__ sum,
                          float* __restrict__ deg, int E) {
    long gid = (long)blockIdx.x * blockDim.x + threadIdx.x;
    int e = (int)(gid >> 5);
    int lane = (int)(gid & 31);
    if (e >= E) return;
    int s = src[e];
    int d = dst[e];
    const float4* sp = (const float4*)(hsrc + (long)s * HIDDEN);
    float4 v = sp[lane];
    float* dp = sum + (long)d * HIDDEN + lane * 4;
    atomicAdd(dp + 0, v.x);
    atomicAdd(dp + 1, v.y);
    atomicAdd(dp + 2, v.z);
    atomicAdd(dp + 3, v.w);
    if (lane == 0) atomicAdd(deg + d, 1.0f);
}

// ---------------- combine (1 aggregate): h = (ALPHA*h + sum/(deg+eps)) / 2 ----------------
__global__ void k_combine2(float* __restrict__ h, const float* __restrict__ sum,
                           const float* __restrict__ deg, long n_elem) {
    long i = (long)blockIdx.x * blockDim.x + threadIdx.x;
    if (i >= n_elem) return;
    long node = i >> 7;
    float agg = sum[i] / (deg[node] + EPS);
    h[i] = (ALPHA * h[i] + agg) * 0.5f;
}

// ---------------- combine (2 aggregates): h = (ALPHA*h + a1 + a2) / 3 ----------------
__global__ void k_combine3(float* __restrict__ h,
                           const float* __restrict__ s1, const float* __restrict__ d1,
                           const float* __restrict__ s2, const float* __restrict__ d2,
                           long n_elem) {
    long i = (long)blockIdx.x * blockDim.x + threadIdx.x;
    if (i >= n_elem) return;
    long node = i >> 7;
    float a1 = s1[i] / (d1[node] + EPS);
    float a2 = s2[i] / (d2[node] + EPS);
    h[i] = (ALPHA * h[i] + a1 + a2) * (1.0f / 3.0f);
}

// ---------------- out[m,o] = h[m,:] . Wout[:,o] + b[o] ----------------
__global__ void k_out(const float* __restrict__ h, const float* __restrict__ Wo,
                      const float* __restrict__ bo, float* __restrict__ out) {
    int idx = blockIdx.x * blockDim.x + threadIdx.x;
    if (idx >= N_PAPER * OUT_DIM) return;
    int m = idx >> 4;
    int o = idx & 15;
    const float* hr = h + (long)m * HIDDEN;
    float acc = bo[o];
#pragma unroll 8
    for (int k = 0; k < HIDDEN; k++) acc += hr[k] * Wo[k * OUT_DIM + o];
    out[idx] = acc;
}

extern "C" void kernel_launch(void* const* d_in, const int* in_sizes, int n_in,
                              void* d_out, int out_size, void* d_ws, size_t ws_size,
                              hipStream_t stream) {
    const float* x_paper  = (const float*)d_in[0];
    const float* x_author = (const float*)d_in[1];
    const float* x_field  = (const float*)d_in[2];
    const float* W_paper[2]  = { (const float*)d_in[3], (const float*)d_in[4] };
    const float* W_author[2] = { (const float*)d_in[5], (const float*)d_in[6] };
    const float* W_field[2]  = { (const float*)d_in[7], (const float*)d_in[8] };
    const float* W_out = (const float*)d_in[9];
    const float* b_out = (const float*)d_in[10];
    const int* src_writes = (const int*)d_in[11];
    const int* dst_writes = (const int*)d_in[12];
    const int* src_wb     = (const int*)d_in[13];
    const int* dst_wb     = (const int*)d_in[14];
    const int* src_cites  = (const int*)d_in[15];
    const int* dst_cites  = (const int*)d_in[16];
    const int* src_topic  = (const int*)d_in[17];
    const int* dst_topic  = (const int*)d_in[18];

    const int E_writes = in_sizes[11];
    const int E_wb     = in_sizes[13];
    const int E_cites  = in_sizes[15];
    const int E_topic  = in_sizes[17];

    // workspace layout (floats)
    float* ws = (float*)d_ws;
    float* hP    = ws;
    float* hA    = hP    + (long)N_PAPER  * HIDDEN;
    float* hF    = hA    + (long)N_AUTHOR * HIDDEN;
    float* sumPW = hF    + (long)N_FIELD  * HIDDEN;
    float* sumPC = sumPW + (long)N_PAPER  * HIDDEN;
    float* sumA  = sumPC + (long)N_PAPER  * HIDDEN;
    float* sumF  = sumA  + (long)N_AUTHOR * HIDDEN;
    float* degPW = sumF  + (long)N_FIELD  * HIDDEN;
    float* degPC = degPW + N_PAPER;
    float* degA  = degPC + N_PAPER;
    float* degF  = degA  + N_AUTHOR;
    const long zeroN = (long)2 * N_PAPER * HIDDEN + (long)N_AUTHOR * HIDDEN +
                       (long)N_FIELD * HIDDEN + 2L * N_PAPER + N_AUTHOR + N_FIELD;

    const int GEMM_GRID_CAP = 2048;
    int gP = N_PAPER  / 16; if (gP > GEMM_GRID_CAP) gP = GEMM_GRID_CAP;
    int gA = N_AUTHOR / 16; if (gA > GEMM_GRID_CAP) gA = GEMM_GRID_CAP;
    int gF = N_FIELD  / 16; if (gF > GEMM_GRID_CAP) gF = GEMM_GRID_CAP;

    for (int l = 0; l < 2; l++) {
        // zero aggregation buffers + degree counters
        k_zero<<<2048, 256, 0, stream>>>(sumPW, zeroN);

        // per-type transform + relu (layer 0: from inputs; layer 1: in place)
        const float* inP = (l == 0) ? x_paper  : hP;
        const float* inA = (l == 0) ? x_author : hA;
        const float* inF = (l == 0) ? x_field  : hF;
        k_gemm_relu<<<gP, 256, 0, stream>>>(inP, W_paper[l],  hP, N_PAPER);
        k_gemm_relu<<<gA, 256, 0, stream>>>(inA, W_author[l], hA, N_AUTHOR);
        k_gemm_relu<<<gF, 256, 0, stream>>>(inF, W_field[l],  hF, N_FIELD);

        // mean-aggregation scatters (wave per edge)
        auto blocks = [](long E) { return (int)((E * 32 + 255) / 256); };
        k_scatter<<<blocks(E_writes), 256, 0, stream>>>(hA, src_writes, dst_writes, sumPW, degPW, E_writes);
        k_scatter<<<blocks(E_cites),  256, 0, stream>>>(hP, src_cites,  dst_cites,  sumPC, degPC, E_cites);
        k_scatter<<<blocks(E_wb),     256, 0, stream>>>(hP, src_wb,     dst_wb,     sumA,  degA,  E_wb);
        k_scatter<<<blocks(E_topic),  256, 0, stream>>>(hP, src_topic,  dst_topic,  sumF,  degF,  E_topic);

        // combine (in place)
        long nP = (long)N_PAPER * HIDDEN;
        long nA = (long)N_AUTHOR * HIDDEN;
        long nF = (long)N_FIELD * HIDDEN;
        k_combine3<<<(int)((nP + 255) / 256), 256, 0, stream>>>(hP, sumPW, degPW, sumPC, degPC, nP);
        k_combine2<<<(int)((nA + 255) / 256), 256, 0, stream>>>(hA, sumA, degA, nA);
        k_combine2<<<(int)((nF + 255) / 256), 256, 0, stream>>>(hF, sumF, degF, nF);
    }

    // final projection
    k_out<<<(N_PAPER * OUT_DIM + 255) / 256, 256, 0, stream>>>(hP, W_out, b_out, (float*)d_out);
}